// Op_46806553592381
// MI455X (gfx1250) — compile-verified
//
#include <hip/hip_runtime.h>

#define D_FEAT   128
#define TILE     1024      // edges staged per LDS tile (TILE*4 bytes per array)
#define BLOCK    256
#define WAVES    (BLOCK / 32)

// ---------------------------------------------------------------- zero output
__global__ void zero_out_kernel(float4* __restrict__ out, int n4) {
    int i = blockIdx.x * blockDim.x + threadIdx.x;
    if (i < n4) out[i] = make_float4(0.f, 0.f, 0.f, 0.f);
}

// ------------------------------------------------- CDNA5 async-to-LDS helpers
__device__ __forceinline__ unsigned lds_addr32(const void* p) {
    // Low 32 bits of a generic pointer to __shared__ == LDS byte offset
    // (flat LDS aperture maps addr[31:0] -> LDS address).
    return (unsigned)(unsigned long long)p;
}

__device__ __forceinline__ void async_ld_b128_nt(unsigned lds_dst, unsigned goff,
                                                 const void* sbase) {
    // GVS mode: mem = SADDR(sgpr64) + VADDR(vgpr32) ; dest = LDS byte addr vgpr
    // NT hint: edge triples are read-once stream -> don't evict L2-resident x/out
    asm volatile("global_load_async_to_lds_b128 %0, %1, %2 th:TH_LOAD_NT"
                 :: "v"(lds_dst), "v"(goff), "s"(sbase)
                 : "memory");
}

__device__ __forceinline__ void wait_async_0() {
#if __has_builtin(__builtin_amdgcn_s_wait_asynccnt)
    __builtin_amdgcn_s_wait_asynccnt(0);
#else
    asm volatile("s_wait_asynccnt 0" ::: "memory");
#endif
}

// guaranteed-native f32 atomic add (global_atomic_add_f32, non-returning)
__device__ __forceinline__ void fatomic_add(float* p, float v) {
    unsafeAtomicAdd(p, v);
}

// ----------------------------------------------------------------- SpMM kernel
__global__ __launch_bounds__(BLOCK)
void spmm_scatter_kernel(const float* __restrict__ x,
                         const int*   __restrict__ adj_rows,
                         const int*   __restrict__ adj_cols,
                         const float* __restrict__ adj_vals,
                         const int*   __restrict__ idx_p,
                         float*       __restrict__ out,
                         int n_edges) {
    __shared__ int   s_rows[2][TILE];
    __shared__ int   s_cols[2][TILE];
    __shared__ float s_vals[2][TILE];

    const int tid  = threadIdx.x;
    const int lane = tid & 31;
    const int wave = tid >> 5;

    // idx is a device scalar selecting which adjacency to use; keep it in SGPRs
    const int idx0 = __builtin_amdgcn_readfirstlane(idx_p[0]);
    const int*   rows = adj_rows + (size_t)idx0 * (size_t)n_edges;
    const int*   cols = adj_cols + (size_t)idx0 * (size_t)n_edges;
    const float* vals = adj_vals + (size_t)idx0 * (size_t)n_edges;

    const int n_tiles = (n_edges + TILE - 1) / TILE;

    int tile = blockIdx.x;
    if (tile >= n_tiles) return;

    // stage tile `t` of the edge triples into LDS buffer `buf`
    // 256 threads x 16B = 4KB = TILE*4 bytes per array -> one b128 per lane/array
    auto stage = [&](int t, int buf) {
        unsigned goff = (unsigned)(t * TILE + tid * 4) * 4u;  // byte offset
        async_ld_b128_nt(lds_addr32(&s_rows[buf][tid * 4]), goff, rows);
        async_ld_b128_nt(lds_addr32(&s_cols[buf][tid * 4]), goff, cols);
        async_ld_b128_nt(lds_addr32(&s_vals[buf][tid * 4]), goff, vals);
    };

    int buf = 0;
    stage(tile, buf);

    for (; tile < n_tiles; tile += gridDim.x) {
        const int next = tile + gridDim.x;

        wait_async_0();       // current buffer's async copies landed in LDS
        __syncthreads();      // make them visible to all waves

        if (next < n_tiles) stage(next, buf ^ 1);  // overlap next stage w/ compute

        const int tile_edges = min(TILE, n_edges - tile * TILE);

        // one wave per edge: 32 lanes x float4 = 128 features
        for (int slot = wave; slot < tile_edges; slot += WAVES) {
            const int   row = s_rows[buf][slot];   // LDS broadcast (same addr/lane)
            const int   col = s_cols[buf][slot];
            const float val = s_vals[buf][slot];

            // prefetch x-row for this wave's next edge (L2-resident gather)
            const int pslot = slot + WAVES;
            if (pslot < tile_edges) {
                const float* px =
                    x + (size_t)s_cols[buf][pslot] * D_FEAT + lane * 4;
                __builtin_prefetch(px, 0, 3);
            }

            const float4 v =
                ((const float4*)(x + (size_t)col * D_FEAT))[lane];  // global_load_b128

            float* o = out + (size_t)row * D_FEAT + lane * 4;
            fatomic_add(o + 0, v.x * val);   // global_atomic_add_f32, no return
            fatomic_add(o + 1, v.y * val);
            fatomic_add(o + 2, v.z * val);
            fatomic_add(o + 3, v.w * val);
        }

        __syncthreads();   // all waves done reading buf before it is restaged
        buf ^= 1;
    }
}

// ------------------------------------------------------------------- launcher
extern "C" void kernel_launch(void* const* d_in, const int* in_sizes, int n_in,
                              void* d_out, int out_size, void* d_ws, size_t ws_size,
                              hipStream_t stream) {
    const float* x        = (const float*)d_in[0];
    const int*   adj_rows = (const int*)  d_in[1];
    const int*   adj_cols = (const int*)  d_in[2];
    const float* adj_vals = (const float*)d_in[3];
    const int*   idx      = (const int*)  d_in[4];

    const int n_edges = in_sizes[1] / 3;          // adj arrays are [K_ADJ=3, N_EDGES]
    float* out = (float*)d_out;

    // zero the output (segment_sum accumulator), vectorized float4
    const int n4 = out_size / 4;
    zero_out_kernel<<<(n4 + BLOCK - 1) / BLOCK, BLOCK, 0, stream>>>(
        (float4*)out, n4);

    // 3125 tiles; ~3 tiles per block -> double-buffered async pipeline engages,
    // 1024 blocks x 8 waves = 8192 waves of latency-hiding parallelism.
    const int n_tiles = (n_edges + TILE - 1) / TILE;
    const int grid = n_tiles < 1024 ? n_tiles : 1024;
    spmm_scatter_kernel<<<grid, BLOCK, 0, stream>>>(
        x, adj_rows, adj_cols, adj_vals, idx, out, n_edges);
}